// _NonLocalBlockND_6081673691369
// MI455X (gfx1250) — compile-verified
//
#include <hip/hip_runtime.h>
#include <hip/hip_bf16.h>

typedef __attribute__((ext_vector_type(2))) float v2f;
typedef __attribute__((ext_vector_type(8))) float v8f;

__device__ __forceinline__ v8f wmma_f32_16x16x4(v2f a, v2f b, v8f c) {
    // D = A(16x4) * B(4x16) + C(16x16), fp32, wave32
    return __builtin_amdgcn_wmma_f32_16x16x4_f32(
        /*neg_a=*/false, a, /*neg_b=*/false, b,
        /*c_mod=*/(short)0, c, /*reuse_a=*/false, /*reuse_b=*/false);
}

#define NSUB 4   // N-subtiles per wave in GEMMs (16x64 strip per wave)

// ---------------------------------------------------------------------------
// Out[b, oc, n] = sum_k Wt[oc,k] * X[b,k,n] + bias[oc]
// One wave (32 threads) per block; each wave computes a 16 x (16*NSUB) tile.
// A-fragment (f32 16x4): lane = M row (lane&15), K pair = 2*(lane>>4)+{0,1}.
// B-fragment (f32 4x16): lane = N col, same K split.
// D (16x16 f32): VGPR v -> M = v + 8*(lane>>4), N = lane&15.
// ---------------------------------------------------------------------------
__global__ __launch_bounds__(32) void conv1x1_wmma(
    const float* __restrict__ Wt, const float* __restrict__ X,
    const float* __restrict__ bias, float* __restrict__ Out,
    int M, int K, int N) {
  const int lane = threadIdx.x & 31;
  const int row  = lane & 15;
  const int half = lane >> 4;
  const int n0 = blockIdx.x * (16 * NSUB);
  const int m0 = blockIdx.y * 16;
  const int b  = blockIdx.z;
  const float* Xb = X + (size_t)b * K * N;
  float* Ob = Out + (size_t)b * M * N;

  v8f acc[NSUB];
  #pragma unroll
  for (int s = 0; s < NSUB; ++s) acc[s] = {};

  const float* Arow = Wt + (size_t)(m0 + row) * K;
  for (int k0 = 0; k0 < K; k0 += 4) {
    const int ka = k0 + 2 * half;                 // this lane's K pair
    v2f afrag = *(const v2f*)(Arow + ka);         // A[row, ka], A[row, ka+1]
    const float* Xk = Xb + (size_t)ka * N;
    #pragma unroll
    for (int s = 0; s < NSUB; ++s) {
      const int n = n0 + 16 * s + row;            // B column = lane row
      v2f bfrag;
      bfrag.x = Xk[n];
      bfrag.y = Xk[N + n];
      acc[s] = wmma_f32_16x16x4(afrag, bfrag, acc[s]);
    }
  }
  #pragma unroll
  for (int s = 0; s < NSUB; ++s) {
    const int n = n0 + 16 * s + row;
    #pragma unroll
    for (int v = 0; v < 8; ++v) {
      const int oc = m0 + v + 8 * half;           // D row mapping
      Ob[(size_t)oc * N + n] = acc[s][v] + bias[oc];
    }
  }
}

// ---------------------------------------------------------------------------
// va[c] = sum_ci w_theta[ci] * theta_w[ci,c];  va[C] = w_theta . theta_b
// (theta conv collapses: a = va . x[b,:,n] + va[C])
// ---------------------------------------------------------------------------
__global__ void make_va(const float* __restrict__ cp_w,
                        const float* __restrict__ theta_w,
                        const float* __restrict__ theta_b,
                        float* __restrict__ va, int Ci, int C) {
  const int c = threadIdx.x;
  float s = 0.f;
  for (int ci = 0; ci < Ci; ++ci) s += cp_w[ci] * theta_w[ci * C + c];
  va[c] = s;
  if (c == 0) {
    float ca = 0.f;
    for (int ci = 0; ci < Ci; ++ci) ca += cp_w[ci] * theta_b[ci];
    va[C] = ca;
  }
}

__global__ void compute_a(const float* __restrict__ x,
                          const float* __restrict__ va,
                          float* __restrict__ avec, int C, int N) {
  __shared__ float sva[257];
  sva[threadIdx.x] = va[threadIdx.x];
  if (threadIdx.x == 0) sva[256] = va[C];
  __syncthreads();
  const int b = blockIdx.y;
  const int n = blockIdx.x * blockDim.x + threadIdx.x;
  const float* xb = x + (size_t)b * C * N;
  float s = sva[256];
  for (int c = 0; c < C; ++c) s += sva[c] * xb[(size_t)c * N + n];
  avec[(size_t)b * N + n] = s;
}

// g_pool[b, m, ci] = 2x2 max of gfull[b, ci, :]  (transposed for GEMM B-matrix)
__global__ void pool_g(const float* __restrict__ gf, float* __restrict__ gp,
                       int Ci, int H, int Wd) {
  const int Wh = Wd >> 1;
  const int Nk = (H >> 1) * Wh;
  const int m  = blockIdx.x * blockDim.x + threadIdx.x;
  const int ci = blockIdx.y;
  const int b  = blockIdx.z;
  const int hh = m / Wh, ww = m % Wh;
  const float* src = gf + ((size_t)b * Ci + ci) * (H * Wd) + (2 * hh) * Wd + 2 * ww;
  const float mx = fmaxf(fmaxf(src[0], src[1]), fmaxf(src[Wd], src[Wd + 1]));
  gp[((size_t)b * Nk + m) * Ci + ci] = mx;
}

// bvec[b,m] = sum_ci w_phi[ci] * maxpool2(phi_full)[b,ci,m]   (fused pool+dot)
__global__ void pool_dot_phi(const float* __restrict__ pf,
                             const float* __restrict__ cp_w,
                             float* __restrict__ bvec, int Ci, int H, int Wd) {
  const int Wh = Wd >> 1;
  const int Nk = (H >> 1) * Wh;
  const int m = blockIdx.x * blockDim.x + threadIdx.x;
  const int b = blockIdx.y;
  const int hh = m / Wh, ww = m % Wh;
  const int p = (2 * hh) * Wd + 2 * ww;
  const float* base = pf + (size_t)b * Ci * H * Wd;
  float s = 0.f;
  for (int ci = 0; ci < Ci; ++ci) {
    const float* src = base + (size_t)ci * H * Wd + p;
    const float mx = fmaxf(fmaxf(src[0], src[1]), fmaxf(src[Wd], src[Wd + 1]));
    s += cp_w[Ci + ci] * mx;
  }
  bvec[(size_t)b * Nk + m] = s;
}

// ---------------------------------------------------------------------------
// Y[b, ci, n] = sum_m relu(a[b,n] + bvec[b,m]) / Nk * gp[b, m, ci]
// f is synthesized per-lane in registers as the WMMA A-fragment.
// ---------------------------------------------------------------------------
__global__ __launch_bounds__(32) void attn_y_wmma(
    const float* __restrict__ avec, const float* __restrict__ bvec,
    const float* __restrict__ gp, float* __restrict__ Y,
    int Ci, int Nq, int Nk) {
  const int lane = threadIdx.x & 31;
  const int row  = lane & 15;
  const int half = lane >> 4;
  const int n0  = blockIdx.x * (16 * NSUB);
  const int ci0 = blockIdx.y * 16;
  const int b   = blockIdx.z;
  const float invNk = 1.0f / (float)Nk;
  const float* ab = avec + (size_t)b * Nq;
  const float* bb = bvec + (size_t)b * Nk;
  const float* gb = gp + (size_t)b * Nk * Ci;

  float arow[NSUB];
  #pragma unroll
  for (int s = 0; s < NSUB; ++s) arow[s] = ab[n0 + 16 * s + row];

  v8f acc[NSUB];
  #pragma unroll
  for (int s = 0; s < NSUB; ++s) acc[s] = {};

  for (int k0 = 0; k0 < Nk; k0 += 4) {
    const int kb = k0 + 2 * half;
    const v2f bm = *(const v2f*)(bb + kb);        // bvec[kb], bvec[kb+1]
    const float* g0 = gb + (size_t)kb * Ci + ci0 + row;
    v2f bfrag;
    bfrag.x = g0[0];
    bfrag.y = g0[Ci];
    #pragma unroll
    for (int s = 0; s < NSUB; ++s) {
      v2f afrag;                                  // f tile on the fly
      afrag.x = fmaxf(arow[s] + bm.x, 0.f) * invNk;
      afrag.y = fmaxf(arow[s] + bm.y, 0.f) * invNk;
      acc[s] = wmma_f32_16x16x4(afrag, bfrag, acc[s]);
    }
  }
  #pragma unroll
  for (int s = 0; s < NSUB; ++s) {
    const int nbase = n0 + 16 * s + 8 * half;
    const size_t orow = ((size_t)b * Ci + (ci0 + row)) * Nq;
    #pragma unroll
    for (int v = 0; v < 8; ++v)
      Y[orow + nbase + v] = acc[s][v];            // store transposed (ci, n)
  }
}

// per-channel sum / sumsq over (B,H,W)
__global__ void bn_stats(const float* __restrict__ wy, float* __restrict__ sums,
                         int C, int HW) {
  const int c = blockIdx.x;
  const int b = blockIdx.y;
  const float* p = wy + ((size_t)b * C + c) * HW;
  float s = 0.f, s2 = 0.f;
  for (int i = threadIdx.x; i < HW; i += blockDim.x) {
    const float v = p[i];
    s += v; s2 += v * v;
  }
  __shared__ float sh0[256], sh1[256];
  sh0[threadIdx.x] = s; sh1[threadIdx.x] = s2;
  __syncthreads();
  for (int off = 128; off > 0; off >>= 1) {
    if ((int)threadIdx.x < off) {
      sh0[threadIdx.x] += sh0[threadIdx.x + off];
      sh1[threadIdx.x] += sh1[threadIdx.x + off];
    }
    __syncthreads();
  }
  if (threadIdx.x == 0) {
    atomicAdd(&sums[c], sh0[0]);
    atomicAdd(&sums[C + c], sh1[0]);
  }
}

// out = (wy - mean) * rsqrt(var+eps) * gamma + beta + x   (in place on d_out)
__global__ void bn_apply(float* __restrict__ out, const float* __restrict__ x,
                         const float* __restrict__ sums,
                         const float* __restrict__ gamma,
                         const float* __restrict__ beta,
                         int C, int HW, int B) {
  const size_t i = (size_t)blockIdx.x * blockDim.x + threadIdx.x;
  const size_t total = (size_t)B * C * HW;
  if (i >= total) return;
  const int c = (int)((i / HW) % C);
  const float cnt = (float)B * (float)HW;
  const float mean = sums[c] / cnt;
  const float var  = sums[C + c] / cnt - mean * mean;
  const float inv  = rsqrtf(var + 1e-5f);
  out[i] = (out[i] - mean) * inv * gamma[c] + beta[c] + x[i];
}

extern "C" void kernel_launch(void* const* d_in, const int* in_sizes, int n_in,
                              void* d_out, int out_size, void* d_ws, size_t ws_size,
                              hipStream_t stream) {
  const float* x       = (const float*)d_in[0];
  const float* g_w     = (const float*)d_in[1];
  const float* g_b     = (const float*)d_in[2];
  const float* theta_w = (const float*)d_in[3];
  const float* theta_b = (const float*)d_in[4];
  const float* phi_w   = (const float*)d_in[5];
  const float* phi_b   = (const float*)d_in[6];
  const float* cp_w    = (const float*)d_in[7];
  const float* W_w     = (const float*)d_in[8];
  const float* W_b     = (const float*)d_in[9];
  const float* bn_g    = (const float*)d_in[10];
  const float* bn_b    = (const float*)d_in[11];
  float* out = (float*)d_out;

  const int Ci = in_sizes[2];           // 128
  const int C  = in_sizes[10];          // 256
  const int H = 64, Wd = 64;
  const int N  = H * Wd;                // 4096
  const int Nk = (H / 2) * (Wd / 2);    // 1024
  const int B  = in_sizes[0] / (C * N); // 8

  // Workspace layout (floats), ~21.2 MB total
  float* ws = (float*)d_ws;
  float* tmp_full = ws;                                   // B*Ci*N
  float* g_pool   = tmp_full + (size_t)B * Ci * N;        // B*Nk*Ci
  float* avec     = g_pool + (size_t)B * Nk * Ci;         // B*N
  float* bvec     = avec + (size_t)B * N;                 // B*Nk
  float* va       = bvec + (size_t)B * Nk;                // C+1
  float* sums     = va + (C + 1);                         // 2*C

  // 1. collapsed theta path
  make_va<<<dim3(1), dim3(C), 0, stream>>>(cp_w, theta_w, theta_b, va, Ci, C);
  compute_a<<<dim3(N / 256, B), dim3(256), 0, stream>>>(x, va, avec, C, N);
  // 2. g conv (WMMA) -> tmp_full; pool -> g_pool[b,m,ci]
  conv1x1_wmma<<<dim3(N / (16 * NSUB), Ci / 16, B), dim3(32), 0, stream>>>(
      g_w, x, g_b, tmp_full, Ci, C, N);
  pool_g<<<dim3(Nk / 256, Ci, B), dim3(256), 0, stream>>>(tmp_full, g_pool, Ci, H, Wd);
  // 3. phi conv (WMMA) -> tmp_full (reused); fused pool+dot -> bvec
  conv1x1_wmma<<<dim3(N / (16 * NSUB), Ci / 16, B), dim3(32), 0, stream>>>(
      phi_w, x, phi_b, tmp_full, Ci, C, N);
  pool_dot_phi<<<dim3(Nk / 256, B), dim3(256), 0, stream>>>(tmp_full, cp_w, bvec, Ci, H, Wd);
  // 4. y = (relu(a+b)/Nk) @ g  (WMMA, f synthesized) -> tmp_full as [b,ci,n]
  attn_y_wmma<<<dim3(N / (16 * NSUB), Ci / 16, B), dim3(32), 0, stream>>>(
      avec, bvec, g_pool, tmp_full, Ci, N, Nk);
  // 5. final conv (WMMA) -> d_out (pre-BN wy)
  conv1x1_wmma<<<dim3(N / (16 * NSUB), C / 16, B), dim3(32), 0, stream>>>(
      W_w, tmp_full, W_b, out, C, Ci, N);
  // 6. batchnorm stats + apply + residual
  hipMemsetAsync(sums, 0, sizeof(float) * 2 * C, stream);
  bn_stats<<<dim3(C, B), dim3(256), 0, stream>>>(out, sums, C, N);
  const size_t total = (size_t)B * C * N;
  bn_apply<<<dim3((unsigned)((total + 255) / 256)), dim3(256), 0, stream>>>(
      out, x, sums, bn_g, bn_b, C, N, B);
  (void)theta_b; (void)n_in; (void)out_size; (void)ws_size;
}